// ResidualQuantization_21878563406303
// MI455X (gfx1250) — compile-verified
//
#include <hip/hip_runtime.h>

typedef __attribute__((ext_vector_type(2))) float v2f;
typedef __attribute__((ext_vector_type(8))) float v8f;

#define NROWS   131072          // B*H*W
#define NTILES  8192            // NROWS/16
#define KCODES  512
#define CDIM    64
#define NQ      4
#define HW      16384           // 128*128
#define ROWSTR  68              // padded LDS codebook row stride (dwords); 68%64=4 -> conflict-free
#define CSQ_OFF (KCODES*ROWSTR)
#define LDS_FLOATS (CSQ_OFF + KCODES)
#define QUANT_ELEMS (8*64*16384)        // 8388608
#define IDX_BASE  QUANT_ELEMS
#define LOSS_BASE (QUANT_ELEMS + NROWS*NQ)   // 8912896

extern "C" __global__ __launch_bounds__(256)
void rvq_wmma_kernel(const float* __restrict__ x,
                     const float* __restrict__ codebook,
                     float* __restrict__ out)
{
  extern __shared__ float smem[];

  // ---- stage codebook into LDS (padded rows), and -0.5*||c||^2 table ----
  for (int i = threadIdx.x; i < KCODES * 16; i += 256) {
    int row = i >> 4, ch = i & 15;
    float4 v = ((const float4*)codebook)[(row << 4) + ch];
    *(float4*)&smem[row * ROWSTR + (ch << 2)] = v;   // byte off = row*272+ch*16, 16B aligned
  }
  for (int c = threadIdx.x; c < KCODES; c += 256) {
    const float* cr = codebook + c * CDIM;
    float s = 0.f;
#pragma unroll
    for (int k = 0; k < CDIM; k += 4) {
      float4 v = *(const float4*)(cr + k);
      s += v.x * v.x + v.y * v.y + v.z * v.z + v.w * v.w;
    }
    smem[CSQ_OFF + c] = -0.5f * s;
  }
  __syncthreads();

  const int lane = threadIdx.x & 31;
  const int wv   = threadIdx.x >> 5;
  const int m    = lane & 15;      // row within tile
  const int hi   = lane >> 4;      // which K-pair / code-octet half
  const int gw   = blockIdx.x * 8 + wv;
  const int nw   = gridDim.x * 8;

  int*   idxout = (int*)out + IDX_BASE;
  float* loss   = out + LOSS_BASE;

  for (int t = gw; t < NTILES; t += nw) {
    const int n0  = t << 4;
    const int row = n0 + m;
    const int b   = row >> 14;        // row / (H*W)
    const int hw  = row & (HW - 1);
    const float* xb = x   + (size_t)b * CDIM * HW + hw;
    float*       ob = out + (size_t)b * CDIM * HW + hw;

    // residual B-fragments: lane holds row m, cols {4k+2*hi, 4k+2*hi+1}
    v2f r[16], o[16];
#pragma unroll
    for (int kc = 0; kc < 16; ++kc) {
      const int col0 = (kc << 2) + (hi << 1);
      v2f v;
      v.x = xb[(size_t)col0 * HW];
      v.y = xb[(size_t)(col0 + 1) * HW];
      r[kc] = v;
      o[kc] = v;
    }

    for (int q = 0; q < NQ; ++q) {
      float bestv = -3.4e38f;
      int   bestc = 0;

      // two code tiles in flight -> two independent WMMA accumulation chains
      for (int tb = 0; tb < 32; tb += 2) {
        v8f acc0, acc1;
        {
          const int cs0 = CSQ_OFF + (tb << 4) + (hi << 3);
          float4 a0 = *(const float4*)&smem[cs0];
          float4 a1 = *(const float4*)&smem[cs0 + 4];
          float4 b0 = *(const float4*)&smem[cs0 + 16];
          float4 b1 = *(const float4*)&smem[cs0 + 20];
          acc0[0] = a0.x; acc0[1] = a0.y; acc0[2] = a0.z; acc0[3] = a0.w;
          acc0[4] = a1.x; acc0[5] = a1.y; acc0[6] = a1.z; acc0[7] = a1.w;
          acc1[0] = b0.x; acc1[1] = b0.y; acc1[2] = b0.z; acc1[3] = b0.w;
          acc1[4] = b1.x; acc1[5] = b1.y; acc1[6] = b1.z; acc1[7] = b1.w;
        }
        const int arow0 = ((tb << 4) + m) * ROWSTR + (hi << 1);
        const int arow1 = arow0 + 16 * ROWSTR;
#pragma unroll
        for (int kc = 0; kc < 16; ++kc) {
          v2f a0 = *(const v2f*)&smem[arow0 + (kc << 2)];
          v2f a1 = *(const v2f*)&smem[arow1 + (kc << 2)];
          acc0 = __builtin_amdgcn_wmma_f32_16x16x4_f32(
                     false, a0, false, r[kc], (short)0, acc0, false, false);
          acc1 = __builtin_amdgcn_wmma_f32_16x16x4_f32(
                     false, a1, false, r[kc], (short)0, acc1, false, false);
        }
#pragma unroll
        for (int j = 0; j < 8; ++j) {
          if (acc0[j] > bestv) { bestv = acc0[j]; bestc = (tb << 4) + (hi << 3) + j; }
        }
#pragma unroll
        for (int j = 0; j < 8; ++j) {
          if (acc1[j] > bestv) { bestv = acc1[j]; bestc = ((tb + 1) << 4) + (hi << 3) + j; }
        }
      }

      // combine the two lane-halves covering the same row; prefer lower index on tie
      float ov = __shfl_xor(bestv, 16, 32);
      int   oc = __shfl_xor(bestc, 16, 32);
      if (ov > bestv || (ov == bestv && oc < bestc)) { bestv = ov; bestc = oc; }

      if (lane < 16) idxout[(size_t)(n0 + lane) * NQ + q] = bestc;

      // residual -= codebook[bestc]; loss_q += ||new residual||^2 / (N*C)
      const int crow = bestc * ROWSTR + (hi << 1);
      float ss = 0.f;
#pragma unroll
      for (int kc = 0; kc < 16; ++kc) {
        v2f cb = *(const v2f*)&smem[crow + (kc << 2)];
        r[kc].x -= cb.x; r[kc].y -= cb.y;
        ss += r[kc].x * r[kc].x + r[kc].y * r[kc].y;
      }
#pragma unroll
      for (int off = 16; off; off >>= 1) ss += __shfl_xor(ss, off, 32);
      if (lane == 0) atomicAdd(&loss[q], ss * (1.0f / 8388608.0f));
    }

    // quantized = x - final residual (straight-through sum == x - residual_final)
#pragma unroll
    for (int kc = 0; kc < 16; ++kc) {
      const int col0 = (kc << 2) + (hi << 1);
      ob[(size_t)col0 * HW]       = o[kc].x - r[kc].x;
      ob[(size_t)(col0 + 1) * HW] = o[kc].y - r[kc].y;
    }
  }
}

extern "C" void kernel_launch(void* const* d_in, const int* in_sizes, int n_in,
                              void* d_out, int out_size, void* d_ws, size_t ws_size,
                              hipStream_t stream) {
  const float* x  = (const float*)d_in[0];
  const float* cb = (const float*)d_in[1];
  float* out = (float*)d_out;

  // zero the commit_loss slots (atomically accumulated)
  hipMemsetAsync(out + LOSS_BASE, 0, NQ * sizeof(float), stream);

  dim3 grid(512), block(256);
  size_t shmem = (size_t)LDS_FLOATS * sizeof(float);   // ~141 KB of the 320 KB WGP LDS
  hipLaunchKernelGGL(rvq_wmma_kernel, grid, block, shmem, stream, x, cb, out);
}